// GraphConvolutionLayer_6657199308987
// MI455X (gfx1250) — compile-verified
//
#include <hip/hip_runtime.h>

// GraphConvolutionLayer for MI455X (gfx1250, wave32)
//   h = segment_sum(x[src], dst)      -- edge scatter with f32 L2 atomics
//   out = h @ W.T + b                 -- V_WMMA_F32_16X16X4_F32 tiles
//
// N=10000, E=640000, D_IN=D_OUT=128. All operands are L2-resident (<16 MB).

#define N_NODES 10000
#define N_EDGES 640000
#define D       128

typedef float v2f __attribute__((ext_vector_type(2)));
typedef float v8f __attribute__((ext_vector_type(8)));

// ---------------------------------------------------------------------------
// Kernel 1: zero the accumulator h in workspace (harness poisons d_ws).
// 10000*128 floats = 320000 float4 stores.
// ---------------------------------------------------------------------------
__global__ void gcl_zero_h(float4* __restrict__ h4) {
    unsigned i = blockIdx.x * blockDim.x + threadIdx.x;
    h4[i] = make_float4(0.f, 0.f, 0.f, 0.f);
}

// ---------------------------------------------------------------------------
// Kernel 2: edge scatter. One wave (32 lanes) per edge; lane c handles floats
// [4c, 4c+4) of the 128-wide feature row. The row read is one coalesced
// global_load_b128 per lane; the accumulate is 4 returnless
// global_atomic_add_f32 (STOREcnt, no wait) into h[dst].
// ---------------------------------------------------------------------------
__global__ void gcl_edge_scatter(const float* __restrict__ x,
                                 const int*   __restrict__ src,
                                 const int*   __restrict__ dst,
                                 float*       __restrict__ h) {
    unsigned gid  = blockIdx.x * blockDim.x + threadIdx.x;
    unsigned e    = gid >> 5;    // edge id (grid sized exactly to 32*E threads)
    unsigned lane = gid & 31u;   // float4 chunk within the row

    int s = src[e];
    int d = dst[e];

    float4 v = ((const float4*)(x + (size_t)s * D))[lane];
    float* hp = h + (size_t)d * D + lane * 4u;

    unsafeAtomicAdd(hp + 0, v.x);
    unsafeAtomicAdd(hp + 1, v.y);
    unsafeAtomicAdd(hp + 2, v.z);
    unsafeAtomicAdd(hp + 3, v.w);
}

// ---------------------------------------------------------------------------
// Kernel 3: out = h @ W.T + b via V_WMMA_F32_16X16X4_F32.
// One wave per 16x16 output tile. Block = 256 threads = 8 waves = the 8
// column tiles of one 16-row stripe; grid = 625 row stripes.
//
// Fragment layouts per CDNA5 ISA 7.12.2 (32-bit, wave32):
//   A 16x4 : lane L -> row M=L%16, VGPR i -> K = (L/16)*2 + i
//   B 4x16 : lane L -> col N=L%16, VGPR i -> K = (L/16)*2 + i
//   C/D    : lane L -> col N=L%16, VGPR i -> row M = i + 8*(L/16)
// B[k][n] = W.T[k][n] = W[n][k] = W[n*128 + k]  (W stored [D_OUT, D_IN]).
// ---------------------------------------------------------------------------
__global__ void __launch_bounds__(256)
gcl_gemm_wmma(const float* __restrict__ h,
              const float* __restrict__ W,
              const float* __restrict__ bias,
              float*       __restrict__ out) {
    const unsigned lane = threadIdx.x & 31u;
    const unsigned wave = threadIdx.x >> 5;       // 0..7 -> column tile
    const unsigned row0 = blockIdx.x * 16u;       // node-row tile
    const unsigned col0 = wave * 16u;

    const unsigned mn   = lane & 15u;             // row for A, col for B/D
    const unsigned kh   = (lane >> 4) * 2u;       // K sub-offset for this half
    const unsigned half = lane >> 4;

    const float* aRow = h + (size_t)(row0 + mn) * D + kh;  // h row (A)
    const float* bRow = W + (size_t)(col0 + mn) * D + kh;  // W row (B = W.T col)

    v8f acc = {};
#pragma unroll
    for (int k = 0; k < D; k += 4) {
        v2f a, b;
        a.x = aRow[k];
        a.y = aRow[k + 1];
        b.x = bRow[k];
        b.y = bRow[k + 1];
        // D = A(16x4) x B(4x16) + C ; 8 args: neg_a, A, neg_b, B, c_mod, C,
        // reuse_a, reuse_b
        acc = __builtin_amdgcn_wmma_f32_16x16x4_f32(
            false, a, false, b, (short)0, acc, false, false);
    }

    const float bb = bias[col0 + mn];             // same N for all 8 rows
    float* orow = out + (size_t)(row0 + 8u * half) * D + col0 + mn;
#pragma unroll
    for (int i = 0; i < 8; ++i)
        orow[(size_t)i * D] = acc[i] + bb;
}

// ---------------------------------------------------------------------------
// Launch: inputs in setup_inputs() order: x, src, dst, W, b (all f32/i32).
// d_ws holds h (5.12 MB required).
// ---------------------------------------------------------------------------
extern "C" void kernel_launch(void* const* d_in, const int* in_sizes, int n_in,
                              void* d_out, int out_size, void* d_ws, size_t ws_size,
                              hipStream_t stream) {
    const float* x   = (const float*)d_in[0];
    const int*   src = (const int*)  d_in[1];
    const int*   dst = (const int*)  d_in[2];
    const float* W   = (const float*)d_in[3];
    const float* b   = (const float*)d_in[4];
    float*       out = (float*)d_out;
    float*       h   = (float*)d_ws;   // N_NODES * D floats = 5.12 MB

    // 1) h = 0
    gcl_zero_h<<<(N_NODES * D / 4) / 256, 256, 0, stream>>>((float4*)h);

    // 2) h[dst] += x[src]  (32 lanes per edge, exact grid: E*32 threads)
    gcl_edge_scatter<<<(N_EDGES * 32) / 256, 256, 0, stream>>>(x, src, dst, h);

    // 3) out = h @ W.T + b  (625 row-stripes x 8 col-tiles per block)
    gcl_gemm_wmma<<<N_NODES / 16, 256, 0, stream>>>(h, W, b, out);
}